// Cluster_56521769615818
// MI455X (gfx1250) — compile-verified
//
#include <hip/hip_runtime.h>

// MI455X / gfx1250, wave32. f32 WMMA (16x16x4) GEMM path for the four
// 96x96x100352 1x1-conv GEMMs; VALU+LDS kernel for the tiny clustering stage.

typedef float v2f __attribute__((ext_vector_type(2)));
typedef float v8f __attribute__((ext_vector_type(8)));

#define CONV_THREADS 256

// One wave computes a 96x16 output tile: C[96,16] = W[96,96] @ X[96,16] + b.
// K=96 -> 24 x v_wmma_f32_16x16x4_f32 per 16x16 row-tile, 6 row-tiles.
// blockIdx.y selects which conv (weights/bias/output) to run so the three
// fused input convs share one kernel with <64KB static LDS each.
__global__ __launch_bounds__(CONV_THREADS) void conv96_wmma(
    const float* __restrict__ X,
    const float* __restrict__ W0, const float* __restrict__ bb0, float* __restrict__ O0,
    const float* __restrict__ W1, const float* __restrict__ bb1, float* __restrict__ O1,
    const float* __restrict__ W2, const float* __restrict__ bb2, float* __restrict__ O2,
    int nColTiles)
{
    __shared__ float sW[96 * 97];   // pitch 97: odd stride -> conflict-free A reads
    __shared__ float sBias[96];

    const int cv = blockIdx.y;
    const float* W  = (cv == 0) ? W0 : (cv == 1 ? W1 : W2);
    const float* Bb = (cv == 0) ? bb0 : (cv == 1 ? bb1 : bb2);
    float* O        = (cv == 0) ? O0 : (cv == 1 ? O1 : O2);

    for (int i = threadIdx.x; i < 96 * 96; i += CONV_THREADS)
        sW[(i / 96) * 97 + (i % 96)] = W[i];
    if (threadIdx.x < 96) sBias[threadIdx.x] = Bb[threadIdx.x];
    __syncthreads();

    const int lane = threadIdx.x & 31;
    const int wave = threadIdx.x >> 5;
    const int mrow = lane & 15;   // M (A rows / D rows) and N (columns) index
    const int hi   = lane >> 4;   // lane half: selects K pair (A/B) and M+8 (C/D)
    const int wavesPerGrid = gridDim.x * (CONV_THREADS / 32);

    for (int t = blockIdx.x * (CONV_THREADS / 32) + wave; t < nColTiles; t += wavesPerGrid) {
        const int bb = t / 196;            // batch
        const int p0 = (t % 196) * 16;     // pixel tile (3136 = 196*16, never crosses b)
        const float* xb = X + (size_t)bb * (96 * 3136) + p0 + mrow;

        // Speculative prefetch of next tile's activations (global_prefetch_b8).
        const int t2 = t + wavesPerGrid;
        if (t2 < nColTiles)
            __builtin_prefetch(X + (size_t)(t2 / 196) * (96 * 3136) + (t2 % 196) * 16 + mrow, 0, 1);

        // B fragments (4x16 f32 each): VGPR0 = B[K=2*hi][N=mrow], VGPR1 = B[K=2*hi+1][N].
        v2f bf[24];
        #pragma unroll
        for (int kk = 0; kk < 24; ++kk) {
            const float* p = xb + (size_t)(kk * 4 + hi * 2) * 3136;
            bf[kk].x = p[0];
            bf[kk].y = p[3136];
        }

        // C/D layout: VGPR v -> M = v + 8*hi, N = mrow. Bias broadcast along N.
        v8f acc[6];
        #pragma unroll
        for (int r = 0; r < 6; ++r)
            #pragma unroll
            for (int v = 0; v < 8; ++v)
                acc[r][v] = sBias[r * 16 + hi * 8 + v];

        #pragma unroll
        for (int r = 0; r < 6; ++r) {
            const float* wrow = &sW[(r * 16 + mrow) * 97];
            #pragma unroll
            for (int kk = 0; kk < 24; ++kk) {
                // A (16x4 f32): lane L holds row M=L%16; VGPR0=K(2*hi), VGPR1=K(2*hi+1).
                v2f a;
                a.x = wrow[kk * 4 + hi * 2];
                a.y = wrow[kk * 4 + hi * 2 + 1];
                acc[r] = __builtin_amdgcn_wmma_f32_16x16x4_f32(
                    false, a, false, bf[kk], (short)0, acc[r], false, false);
            }
        }

        float* ob = O + (size_t)bb * (96 * 3136) + p0 + mrow;
        #pragma unroll
        for (int r = 0; r < 6; ++r)
            #pragma unroll
            for (int v = 0; v < 8; ++v)
                ob[(size_t)(r * 16 + hi * 8 + v) * 3136] = acc[r][v];
    }
}

// One workgroup per window-batch b' (512 total). All reshapes in the reference
// are flat reinterprets: value/feature chunks are [784,24] row-major, xh chunk
// is [24,784]. Everything here is L2-resident (3 x 38MB << 192MB L2).
__global__ __launch_bounds__(256) void cluster96(
    const float* __restrict__ Vbuf, const float* __restrict__ Fbuf,
    const float* __restrict__ Xbuf, const float* __restrict__ Wc,
    const float* __restrict__ bc, const float* __restrict__ alphap,
    const float* __restrict__ betap, float* __restrict__ Ybuf)
{
    const size_t base = (size_t)blockIdx.x * 18816;
    const float* V = Vbuf + base;   // [n][c], n<784, c<24
    const float* F = Fbuf + base;   // [n][c] (and [c][n] view for pooling)
    const float* X = Xbuf + base;   // [c][n]
    float* Y = Ybuf + base;

    __shared__ float s_cen0[96];     // pooled xh, flat [c*4+m]
    __shared__ float s_cf[96];       // pooled feature, flat == centers_feature[m][c] at m*24+c
    __shared__ float s_C1[96];       // Wc-conv centers, flat [o*4+m] == C1[mm][cc] at mm*24+cc
    __shared__ float s_s0[4][784];   // softmax attention
    __shared__ float s_SF[96];       // sim0 @ feature, [m][c]
    __shared__ float s_cnn[96];      // normalized centers, [m][c]
    __shared__ float s_sval[784];
    __shared__ int   s_widx[784];
    __shared__ float s_outC[96];
    __shared__ int   s_cnt[4];
    __shared__ float s_part[2][96];

    const int tid = threadIdx.x;
    const float alpha = alphap[0];
    const float beta  = betap[0];

    // ---- adaptive 2x2 avg-pool on xh and feature ([c][n] views, 14x14 blocks) ----
    if (tid < 192) {
        const int which = tid >= 96;
        const int k = tid % 96;
        const int c = k >> 2, m = k & 3;
        const int i = m >> 1, j = m & 1;
        const float* src = which ? F : X;
        float s = 0.f;
        const int off0 = c * 784 + (i * 14) * 28 + j * 14;
        for (int di = 0; di < 14; ++di) {
            const float* row = src + off0 + di * 28;
            #pragma unroll
            for (int dj = 0; dj < 14; ++dj) s += row[dj];
        }
        if (which) s_cf[k] = s * (1.0f / 196.0f);
        else       s_cen0[k] = s * (1.0f / 196.0f);
    }
    if (tid < 4) s_cnt[tid] = 0;
    __syncthreads();

    // ---- centers = conv1x1(centers, Wc, bc), stored flat [o*4+m] ----
    if (tid < 96) {
        const int o = tid >> 2, m = tid & 3;
        float s = bc[o];
        #pragma unroll
        for (int c = 0; c < 24; ++c) s += Wc[o * 24 + c] * s_cen0[c * 4 + m];
        s_C1[tid] = s;
    }
    __syncthreads();

    // ---- sim0 = softmax over centers of (C1 @ V^T), [4][784] ----
    for (int n = tid; n < 784; n += 256) {
        float vr[24];
        #pragma unroll
        for (int c = 0; c < 24; ++c) vr[c] = V[n * 24 + c];
        float d[4];
        #pragma unroll
        for (int m = 0; m < 4; ++m) {
            float s = 0.f;
            #pragma unroll
            for (int c = 0; c < 24; ++c) s += s_C1[m * 24 + c] * vr[c];
            d[m] = s;
        }
        float mx = fmaxf(fmaxf(d[0], d[1]), fmaxf(d[2], d[3]));
        float e[4], sum = 0.f;
        #pragma unroll
        for (int m = 0; m < 4; ++m) { e[m] = expf(d[m] - mx); sum += e[m]; }
        const float inv = 1.f / sum;
        #pragma unroll
        for (int m = 0; m < 4; ++m) s_s0[m][n] = e[m] * inv;
    }
    __syncthreads();

    // ---- SF = sim0 @ feature : [m][c], split over token halves ----
    if (tid < 192) {
        const int half = tid / 96, k = tid % 96;
        const int m = k / 24, c = k % 24;
        float s = 0.f;
        const int n0 = half * 392;
        for (int n = n0; n < n0 + 392; ++n) s += s_s0[m][n] * F[n * 24 + c];
        s_part[half][k] = s;
    }
    __syncthreads();
    if (tid < 96) s_SF[tid] = s_part[0][tid] + s_part[1][tid];
    __syncthreads();

    // ---- normalized centers: cn[m][c] = SF_flat[c*4+m] / ||.|| ----
    if (tid < 4) {
        const int m = tid;
        float nr = 0.f;
        #pragma unroll
        for (int c = 0; c < 24; ++c) { const float v = s_SF[c * 4 + m]; nr += v * v; }
        const float inv = 1.f / fmaxf(sqrtf(nr), 1e-12f);
        #pragma unroll
        for (int c = 0; c < 24; ++c) s_cnn[m * 24 + c] = s_SF[c * 4 + m] * inv;
    }
    __syncthreads();

    // ---- cosine sim -> sigmoid -> first-argmax over 4 centers ----
    for (int n = tid; n < 784; n += 256) {
        float xr[24], nr = 0.f;
        #pragma unroll
        for (int c = 0; c < 24; ++c) { const float v = X[c * 784 + n]; xr[c] = v; nr += v * v; }
        const float inv = 1.f / fmaxf(sqrtf(nr), 1e-12f);
        float best = -1.f; int bm = 0;
        #pragma unroll
        for (int m = 0; m < 4; ++m) {
            float s = 0.f;
            #pragma unroll
            for (int c = 0; c < 24; ++c) s += s_cnn[m * 24 + c] * xr[c];
            s = 1.f / (1.f + expf(-(beta + alpha * (s * inv))));
            if (s > best) { best = s; bm = m; }   // strict > keeps first max (jnp.argmax)
        }
        s_sval[n] = best; s_widx[n] = bm;
        atomicAdd(&s_cnt[bm], 1);                  // ds_add_u32
    }
    __syncthreads();

    // ---- outC[m][c] = (sum_{n in m} sval*F[n][c] + CF[m][c]) / (cnt_m + 1) ----
    if (tid < 192) {
        const int half = tid / 96, k = tid % 96;
        const int m = k / 24, c = k % 24;
        float s = 0.f;
        const int n0 = half * 392;
        for (int n = n0; n < n0 + 392; ++n)
            if (s_widx[n] == m) s += s_sval[n] * F[n * 24 + c];
        s_part[half][k] = s;
    }
    __syncthreads();
    if (tid < 96) {
        const int m = tid / 24;
        s_outC[tid] = (s_part[0][tid] + s_part[1][tid] + s_cf[tid])
                      / ((float)s_cnt[m] + 1.0f);
    }
    __syncthreads();

    // ---- broadcast winning center row back to tokens (coalesced [n][c] store) ----
    for (int idx = tid; idx < 18816; idx += 256) {
        const int n = idx / 24, c = idx % 24;
        Y[idx] = s_outC[s_widx[n] * 24 + c] * s_sval[n];
    }
}

extern "C" void kernel_launch(void* const* d_in, const int* in_sizes, int n_in,
                              void* d_out, int out_size, void* d_ws, size_t ws_size,
                              hipStream_t stream) {
    (void)in_sizes; (void)n_in; (void)out_size; (void)ws_size;
    const float* x   = (const float*)d_in[0];
    const float* Wv  = (const float*)d_in[1];
    const float* bv  = (const float*)d_in[2];
    const float* Wf  = (const float*)d_in[3];
    const float* bf  = (const float*)d_in[4];
    const float* W1  = (const float*)d_in[5];
    const float* b1  = (const float*)d_in[6];
    const float* Wc  = (const float*)d_in[7];
    const float* bc  = (const float*)d_in[8];
    const float* W2  = (const float*)d_in[9];
    const float* b2  = (const float*)d_in[10];
    const float* al  = (const float*)d_in[11];
    const float* be  = (const float*)d_in[12];
    float* out = (float*)d_out;

    const size_t N = 32ull * 96 * 3136;    // 9,633,792 floats per tensor
    float* ws  = (float*)d_ws;
    float* val = ws;
    float* fea = ws + N;
    float* xh  = ws + 2 * N;
    float* Yb  = ws + 3 * N;

    // Stage A: three fused 1x1 convs (blockIdx.y selects conv). 6272 column
    // tiles; 392 blocks * 8 waves -> 2 tiles per wave.
    dim3 gA(392, 3);
    conv96_wmma<<<gA, CONV_THREADS, 0, stream>>>(x, Wv, bv, val, Wf, bf, fea, W1, b1, xh, 6272);

    // Stage B: clustering, one workgroup per window-batch chunk.
    cluster96<<<512, 256, 0, stream>>>(val, fea, xh, Wc, bc, al, be, Yb);

    // Stage C: final 1x1 conv to d_out.
    dim3 gC(392, 1);
    conv96_wmma<<<gC, CONV_THREADS, 0, stream>>>(Yb, W2, b2, out, W2, b2, out, W2, b2, out, 6272);
}